// Regression_55791625175653
// MI455X (gfx1250) — compile-verified
//
#include <hip/hip_runtime.h>
#include <math.h>

typedef float v2f __attribute__((ext_vector_type(2)));
typedef float v8f __attribute__((ext_vector_type(8)));

#define DC 48
#define HC 64
#define WC 128
#define HF 256
#define WF 512

// 1-D nearest-4x-upsample + ones(4) box (pad lo=1, hi=2) collapses to two
// coarse taps: fine f = 4k+r touches coarse (lo, lo+1) with weights (w0, w1).
__device__ __forceinline__ void up_weights(int f, int& lo, float& w0, float& w1) {
    int r = f & 3, k = f >> 2;
    if (r == 0)      { lo = k - 1; w0 = 1.f; w1 = 3.f; }
    else if (r == 1) { lo = k;     w0 = 4.f; w1 = 0.f; }
    else if (r == 2) { lo = k;     w0 = 3.f; w1 = 1.f; }
    else             { lo = k;     w0 = 2.f; w1 = 2.f; }
}

__global__ __launch_bounds__(256) void regress_wmma_kernel(
    const float* __restrict__ c3, float* __restrict__ out) {
    // Per-wave slice of the H/W-filtered coarse volume: C[48 coarse d][16 px]
    __shared__ float lds[8][DC * 16];

    const int lane  = threadIdx.x & 31;
    const int wave  = threadIdx.x >> 5;
    const int tile  = blockIdx.x * 8 + wave;      // 8192 tiles total, exact
    const int h     = tile >> 5;                  // 512/16 = 32 tiles per row
    const int wbase = (tile & 31) << 4;
    const int p     = lane & 15;                  // pixel within tile / M row
    const int hh    = lane >> 4;                  // half-wave: K split

    // ---------------- Stage 1: H/W box into LDS ----------------
    int wlo; float a0, a1;
    up_weights(wbase + p, wlo, a0, a1);
    int whi = wlo + 1;
    if (wlo < 0)      { wlo = 0; a0 = a1; a1 = 0.f; whi = 1; }
    if (whi > WC - 1) { whi = WC - 1; a1 = 0.f; }

    int hlo; float b0, b1;
    up_weights(h, hlo, b0, b1);
    int hhi = hlo + 1;
    if (hlo < 0)      { hlo = 0; b0 = b1; b1 = 0.f; hhi = 1; }
    if (hhi > HC - 1) { hhi = HC - 1; b1 = 0.f; }

    const float w00 = b0 * a0, w01 = b0 * a1, w10 = b1 * a0, w11 = b1 * a1;
    for (int cd = hh; cd < DC; cd += 2) {
        const float* base = c3 + (size_t)cd * (HC * WC);
        float v = w00 * base[hlo * WC + wlo]
                + w01 * base[hlo * WC + whi]
                + w10 * base[hhi * WC + wlo]
                + w11 * base[hhi * WC + whi];
        lds[wave][cd * 16 + p] = v;
    }
    __syncthreads();

    // ---- Stage 2: depth matmul S = W(192x48) x C(48x16) via WMMA f32 ----
    // D-tile layout: lane n = pixel n, VGPR g + 8*(lane>=16) = depth row.
    // W's nonzero K-steps for M-tile t are only {t-1, t, t+1}.
    float mrun = -INFINITY, srun = 0.f, trun = 0.f;
    const float* cw = lds[wave];
    for (int t = 0; t < 12; ++t) {
        int d = 16 * t + p;                 // this lane's A row (fine depth)
        int lo; float wd0, wd1;
        up_weights(d, lo, wd0, wd1);        // OOB cols never match -> auto clip
        v8f acc = {0.f, 0.f, 0.f, 0.f, 0.f, 0.f, 0.f, 0.f};
        const int kk0 = (t == 0)  ? 0  : t - 1;
        const int kk1 = (t == 11) ? 11 : t + 1;
        for (int kk = kk0; kk <= kk1; ++kk) {
            const int c0 = 4 * kk + 2 * hh; // A/B K layout: halves split K 0,1|2,3
            const int c1 = c0 + 1;
            v2f A, B;
            A.x = (c0 == lo) ? wd0 : ((c0 == lo + 1) ? wd1 : 0.f);
            A.y = (c1 == lo) ? wd0 : ((c1 == lo + 1) ? wd1 : 0.f);
            B.x = cw[c0 * 16 + p];
            B.y = cw[c1 * 16 + p];
            acc = __builtin_amdgcn_wmma_f32_16x16x4_f32(
                false, A, false, B, (short)0, acc, false, false);
        }
        // Fold 8 depth rows (d = 16t + 8*hh + g) into online softmax state.
        float dbase = (float)(16 * t + 8 * hh);
        float lmax = acc[0];
#pragma unroll
        for (int g = 1; g < 8; ++g) lmax = fmaxf(lmax, acc[g]);
        float newm  = fmaxf(mrun, lmax);
        float scale = __expf(mrun - newm);
        srun *= scale; trun *= scale;
#pragma unroll
        for (int g = 0; g < 8; ++g) {
            float e = __expf(acc[g] - newm);
            srun += e;
            trun += e * (dbase + (float)g);
        }
        mrun = newm;
    }

    // Merge the two depth halves of each pixel (lane n <-> lane n+16).
    float m2 = __shfl_xor(mrun, 16, 32);
    float s2 = __shfl_xor(srun, 16, 32);
    float t2 = __shfl_xor(trun, 16, 32);
    float M  = fmaxf(mrun, m2);
    float S  = srun * __expf(mrun - M) + s2 * __expf(m2 - M);
    float T  = trun * __expf(mrun - M) + t2 * __expf(m2 - M);
    if (lane < 16) out[h * WF + wbase + p] = T / S;
}

extern "C" void kernel_launch(void* const* d_in, const int* in_sizes, int n_in,
                              void* d_out, int out_size, void* d_ws, size_t ws_size,
                              hipStream_t stream) {
    (void)in_sizes; (void)n_in; (void)out_size; (void)d_ws; (void)ws_size;
    const float* c3 = (const float*)d_in[2];   // inputs: c1, c2, c3
    float* out = (float*)d_out;                 // [256*512] f32
    // 8192 pixel-tiles / 8 waves per 256-thread block = 1024 blocks (exact).
    regress_wmma_kernel<<<dim3(1024), dim3(256), 0, stream>>>(c3, out);
}